// GCN_BiLSTM_Attn_Temporal_88871463289366
// MI455X (gfx1250) — compile-verified
//
#include <hip/hip_runtime.h>
#include <math.h>

typedef __attribute__((ext_vector_type(2))) float v2f;
typedef __attribute__((ext_vector_type(8))) float v8f;

#define N_NODES 100000
#define N_EDGES 1600000
#define BATCH   256
#define TSEQ    200
#define EMBD    128
#define GHID    64
#define HID     128
#define G4      512     // 4*H
#define DURD    32
#define TDIMN   4
#define INF1    134     // 2 + EMB + TDIM
#define DIN     160     // EMB + DUR

static __device__ __forceinline__ float sigmoidf_(float x) { return 1.0f / (1.0f + expf(-x)); }

static __device__ __forceinline__ v8f wmma4(v2f a, v2f b, v8f c) {
  return __builtin_amdgcn_wmma_f32_16x16x4_f32(false, a, false, b, (short)0, c, false, false);
}

// ---------------- GCN ----------------

__global__ void k_init_deg(float* deg) {
  int i = blockIdx.x * 256 + threadIdx.x;
  if (i < N_NODES) deg[i] = 1.0f;   // self-loop weight (reference uses ones)
}

__global__ void k_deg_scatter(const int* ei, const float* ew, float* deg) {
  int e = blockIdx.x * 256 + threadIdx.x;
  if (e < N_EDGES) atomicAdd(&deg[ei[N_EDGES + e]], ew[e]);
}

__global__ void k_dis(const float* deg, float* dis) {
  int i = blockIdx.x * 256 + threadIdx.x;
  if (i < N_NODES) {
    float d = deg[i];
    dis[i] = (d > 0.0f) ? rsqrtf(fmaxf(d, 1e-12f)) : 0.0f;
  }
}

// xw1[n][j] = feats(n,:) . W1[:,j],  feats = [coords(2) | emb(128) | temporal(4)]
__global__ void k_xw1(const float* __restrict__ xc, const float* __restrict__ emb,
                      const float* __restrict__ tf, const float* __restrict__ W,
                      float* __restrict__ xw) {
  int t = blockIdx.x * 256 + threadIdx.x;
  int n = t >> 6, j = t & 63;
  if (n >= N_NODES) return;
  float acc = xc[n * 2 + 0] * W[0 * GHID + j] + xc[n * 2 + 1] * W[1 * GHID + j];
  for (int d = 0; d < EMBD; ++d)  acc += emb[n * EMBD + d] * W[(2 + d) * GHID + j];
  for (int d = 0; d < TDIMN; ++d) acc += tf[n * TDIMN + d] * W[(2 + EMBD + d) * GHID + j];
  xw[t] = acc;
}

// out = dis[n]^2 * xw[n] + bias   (self-loop + bias init)
__global__ void k_agg_init(const float* __restrict__ dis, const float* __restrict__ xw,
                           const float* __restrict__ bias, float* __restrict__ out,
                           int logC, int total) {
  int t = blockIdx.x * 256 + threadIdx.x;
  if (t >= total) return;
  int n = t >> logC, j = t & ((1 << logC) - 1);
  float d = dis[n];
  out[t] = d * d * xw[t] + bias[j];
}

__global__ void k_agg_scatter(const int* __restrict__ ei, const float* __restrict__ ew,
                              const float* __restrict__ dis, const float* __restrict__ xw,
                              float* __restrict__ out, int logC) {
  int t = blockIdx.x * 256 + threadIdx.x;
  int C = 1 << logC;
  int e = t >> logC, j = t & (C - 1);
  if (e >= N_EDGES) return;
  int r = ei[e], c = ei[N_EDGES + e];
  float nrm = dis[r] * ew[e] * dis[c];
  atomicAdd(&out[c * C + j], nrm * xw[r * C + j]);
}

// xw2[n][j] = relu(z1(n,:)) . W2[:,j]
__global__ void k_xw2(const float* __restrict__ z1, const float* __restrict__ W,
                      float* __restrict__ xw) {
  int t = blockIdx.x * 256 + threadIdx.x;
  int n = t >> 7, j = t & 127;
  if (n >= N_NODES) return;
  float acc = 0.0f;
  for (int k = 0; k < GHID; ++k) acc += fmaxf(z1[n * GHID + k], 0.0f) * W[k * EMBD + j];
  xw[t] = acc;
}

// ---------------- sequence feature build ----------------

__global__ void k_build_x(const float* __restrict__ z, const int* __restrict__ seq,
                          const float* __restrict__ dur,
                          const float* __restrict__ W1, const float* __restrict__ b1,
                          const float* __restrict__ W2, const float* __restrict__ b2,
                          float* __restrict__ X) {
  int bt = blockIdx.x, tid = threadIdx.x;   // blockDim = 160 (5 waves)
  int s = seq[bt];
  float* xr = X + (long long)bt * DIN;
  if (tid < EMBD) {
    xr[tid] = z[(long long)s * EMBD + tid];
  } else {
    int j2 = tid - EMBD;                    // 0..31
    float d1 = log1pf(dur[bt]);
    float acc = b2[j2];
    for (int j = 0; j < 16; ++j) {
      float h = fmaxf(d1 * W1[j] + b1[j], 0.0f);
      acc += h * W2[j * DURD + j2];
    }
    xr[tid] = acc;
  }
}

// Wih (512,160) -> WihT (160,512)
__global__ void k_transpose_wih(const float* __restrict__ W, float* __restrict__ WT) {
  int t = blockIdx.x * 256 + threadIdx.x;
  if (t >= G4 * DIN) return;
  int g = t / DIN, k = t % DIN;
  WT[k * G4 + g] = W[t];
}

// ---------------- WMMA input projection: Xg = X @ WihT + bias ----------------
// grid (4, 3200), block 256 (8 waves); wave -> one 16x16 tile, K=160
__global__ void k_xg(const float* __restrict__ X, const float* __restrict__ WT,
                     const float* __restrict__ bias, float* __restrict__ Xg) {
  __shared__ float Xs[16 * 164];            // stride 164: conflict-free A reads
  int tid = threadIdx.x;
  int m0 = blockIdx.y * 16;
  for (int i = tid; i < 16 * DIN; i += 256) {
    int r = i / DIN, c = i % DIN;
    Xs[r * 164 + c] = X[(long long)(m0 + r) * DIN + c];
  }
  __syncthreads();
  int wave = tid >> 5, lane = tid & 31, lo = lane & 15, hi = lane >> 4;
  int col0 = (blockIdx.x * 8 + wave) * 16;
  v8f acc = {0.f, 0.f, 0.f, 0.f, 0.f, 0.f, 0.f, 0.f};
  for (int k0 = 0; k0 < DIN; k0 += 4) {
    int ka = k0 + hi * 2;
    v2f a, b;
    a.x = Xs[lo * 164 + ka];
    a.y = Xs[lo * 164 + ka + 1];
    b.x = WT[(long long)ka * G4 + col0 + lo];
    b.y = WT[(long long)(ka + 1) * G4 + col0 + lo];
    acc = wmma4(a, b, acc);
  }
  float bi = bias[col0 + lo];
  for (int j = 0; j < 8; ++j) {
    int m = m0 + j + 8 * hi;
    Xg[(long long)m * G4 + col0 + lo] = acc[j] + bi;
  }
}

// ---------------- LSTM recurrence (one block = 16 batch rows, loops T) ----------------
// LDS: WhhT[128][513] + G[16][512] + H[16][132] + C[16][128]  = 312064 B (<320KB WGP LDS)
#define LSTM_LDS_FLOATS (128 * 513 + 16 * 512 + 16 * 132 + 16 * 128)

__global__ void k_lstm(const float* __restrict__ Xg, const float* __restrict__ Whh,
                       const int* __restrict__ lens, float* __restrict__ out, int backward) {
  extern __shared__ float sm[];
  float* Wl = sm;                       // 128*513 (stride 513: conflict-free B reads)
  float* Gl = sm + 128 * 513;           // 16*512
  float* Hl = Gl + 16 * 512;            // 16*132 (stride 132: conflict-free A reads)
  float* Cl = Hl + 16 * 132;            // 16*128
  int tid = threadIdx.x;
  int b0 = blockIdx.x * 16;

  // stage WhhT into LDS (read Whh coalesced once from HBM)
  for (int i = tid; i < G4 * HID; i += 256) {
    int g = i >> 7, k = i & 127;
    Wl[k * 513 + g] = Whh[i];
  }
  for (int i = tid; i < 16 * HID; i += 256) {
    int m = i >> 7, j = i & 127;
    Hl[m * 132 + j] = 0.0f;
    Cl[i] = 0.0f;
  }
  __syncthreads();

  int wave = tid >> 5, lane = tid & 31, lo = lane & 15, hi = lane >> 4;

  for (int s = 0; s < TSEQ; ++s) {
    // G[16x512] = H[16x128] @ WhhT[128x512]   (wave: 4 N-tiles, K=128 -> 32 WMMA each)
    for (int nt = 0; nt < 4; ++nt) {
      int col0 = (wave * 4 + nt) * 16;
      v8f acc = {0.f, 0.f, 0.f, 0.f, 0.f, 0.f, 0.f, 0.f};
      for (int k0 = 0; k0 < HID; k0 += 4) {
        int ka = k0 + hi * 2;
        v2f a, b;
        a.x = Hl[lo * 132 + ka];
        a.y = Hl[lo * 132 + ka + 1];
        b.x = Wl[ka * 513 + col0 + lo];
        b.y = Wl[(ka + 1) * 513 + col0 + lo];
        acc = wmma4(a, b, acc);
      }
      for (int j = 0; j < 8; ++j) Gl[(j + 8 * hi) * 512 + col0 + lo] = acc[j];
    }
    __syncthreads();

    // elementwise gate update: 16*128 = 2048 cells, 8 per thread
    for (int e = tid; e < 16 * HID; e += 256) {
      int m = e >> 7, j = e & 127;
      int bb = b0 + m;
      int len = lens[bb];
      if (s < len) {
        int tx = backward ? (len - 1 - s) : s;
        const float* xr = Xg + ((long long)bb * TSEQ + tx) * G4;
        float gi = Gl[m * 512 + j]       + xr[j];
        float gf = Gl[m * 512 + 128 + j] + xr[128 + j];
        float gg = Gl[m * 512 + 256 + j] + xr[256 + j];
        float go = Gl[m * 512 + 384 + j] + xr[384 + j];
        float c  = Cl[m * 128 + j];
        float cn = sigmoidf_(gf) * c + sigmoidf_(gi) * tanhf(gg);
        float hn = sigmoidf_(go) * tanhf(cn);
        Cl[m * 128 + j] = cn;
        Hl[m * 132 + j] = hn;
        out[((long long)bb * TSEQ + tx) * HID + j] = hn;
      } else {
        out[((long long)bb * TSEQ + s) * HID + j] = 0.0f;
      }
    }
    __syncthreads();
  }
}

// ---------------- attention pooling ----------------

__global__ void k_attn(const float* __restrict__ outF, const float* __restrict__ outB,
                       const float* __restrict__ aw, const float* __restrict__ ab,
                       const int* __restrict__ lens, float* __restrict__ ctx) {
  __shared__ float sc[TSEQ];
  __shared__ float wv[TSEQ];
  __shared__ float red[2];
  int b = blockIdx.x, tid = threadIdx.x;
  int len = lens[b];
  float ninf = -__builtin_inff();
  if (tid < TSEQ) {
    float s = ab[0];
    const float* f  = outF + ((long long)b * TSEQ + tid) * HID;
    const float* bk = outB + ((long long)b * TSEQ + tid) * HID;
    for (int k = 0; k < HID; ++k) s += f[k] * aw[k] + bk[k] * aw[HID + k];
    sc[tid] = (tid < len) ? s : ninf;
  }
  __syncthreads();
  if (tid == 0) {
    float mx = ninf;
    for (int t = 0; t < TSEQ; ++t) mx = fmaxf(mx, sc[t]);
    red[0] = mx;
  }
  __syncthreads();
  if (tid < TSEQ) wv[tid] = expf(sc[tid] - red[0]);
  __syncthreads();
  if (tid == 0) {
    float sum = 0.0f;
    for (int t = 0; t < TSEQ; ++t) sum += wv[t];
    red[1] = 1.0f / sum;
  }
  __syncthreads();
  float inv = red[1];
  float acc = 0.0f;
  if (tid < HID) {
    for (int t = 0; t < TSEQ; ++t) acc += wv[t] * outF[((long long)b * TSEQ + t) * HID + tid];
  } else {
    int k = tid - HID;
    for (int t = 0; t < TSEQ; ++t) acc += wv[t] * outB[((long long)b * TSEQ + t) * HID + k];
  }
  ctx[b * 256 + tid] = acc * inv;
}

// ---------------- final FC: out(256 x 100000) = ctx @ fc_W + fc_b ----------------
// grid (782, 16), block 256; wave -> one 16x16 tile, K=256
__global__ void k_fc(const float* __restrict__ ctx, const float* __restrict__ W,
                     const float* __restrict__ bias, float* __restrict__ out) {
  __shared__ float As[16 * 260];        // stride 260: conflict-free A reads
  int tid = threadIdx.x;
  int m0 = blockIdx.y * 16;
  for (int i = tid; i < 16 * 256; i += 256) {
    int r = i >> 8, c = i & 255;
    As[r * 260 + c] = ctx[(m0 + r) * 256 + c];
  }
  __syncthreads();
  int wave = tid >> 5, lane = tid & 31, lo = lane & 15, hi = lane >> 4;
  int nt = blockIdx.x * 8 + wave;
  if (nt >= (N_NODES / 16)) return;     // wave-uniform guard (EXEC stays all-ones in WMMA)
  int n0 = nt * 16;
  v8f acc = {0.f, 0.f, 0.f, 0.f, 0.f, 0.f, 0.f, 0.f};
  for (int k0 = 0; k0 < 256; k0 += 4) {
    int ka = k0 + hi * 2;
    v2f a, b;
    a.x = As[lo * 260 + ka];
    a.y = As[lo * 260 + ka + 1];
    b.x = W[(long long)ka * N_NODES + n0 + lo];
    b.y = W[(long long)(ka + 1) * N_NODES + n0 + lo];
    acc = wmma4(a, b, acc);
  }
  float bi = bias[n0 + lo];
  for (int j = 0; j < 8; ++j)
    out[(long long)(m0 + j + 8 * hi) * N_NODES + n0 + lo] = acc[j] + bi;
}

// ---------------- launcher ----------------

extern "C" void kernel_launch(void* const* d_in, const int* in_sizes, int n_in,
                              void* d_out, int out_size, void* d_ws, size_t ws_size,
                              hipStream_t stream) {
  const float* xc   = (const float*)d_in[0];
  const float* tf   = (const float*)d_in[1];
  const int*   ei   = (const int*)  d_in[2];
  const float* ew   = (const float*)d_in[3];
  const int*   seq  = (const int*)  d_in[4];
  const float* dur  = (const float*)d_in[5];
  const int*   lens = (const int*)  d_in[6];
  const float* emb  = (const float*)d_in[7];
  const float* W1   = (const float*)d_in[8];
  const float* b1   = (const float*)d_in[9];
  const float* W2   = (const float*)d_in[10];
  const float* b2   = (const float*)d_in[11];
  const float* dW1  = (const float*)d_in[12];
  const float* db1  = (const float*)d_in[13];
  const float* dW2  = (const float*)d_in[14];
  const float* db2  = (const float*)d_in[15];
  const float* WihF = (const float*)d_in[16];
  const float* WhhF = (const float*)d_in[17];
  const float* bF   = (const float*)d_in[18];
  const float* WihB = (const float*)d_in[19];
  const float* WhhB = (const float*)d_in[20];
  const float* bB   = (const float*)d_in[21];
  const float* attW = (const float*)d_in[22];
  const float* attB = (const float*)d_in[23];
  const float* fcW  = (const float*)d_in[24];
  const float* fcb  = (const float*)d_in[25];
  float* out = (float*)d_out;

  float* ws = (float*)d_ws;
  size_t off = 0;
  auto alloc = [&](size_t n) { float* p = ws + off; off += n; return p; };
  float* deg = alloc(N_NODES);
  float* dis = alloc(N_NODES);
  float* xw1 = alloc((size_t)N_NODES * GHID);
  float* o1  = alloc((size_t)N_NODES * GHID);
  float* xw2 = alloc((size_t)N_NODES * EMBD);
  float* z   = alloc((size_t)N_NODES * EMBD);
  float* X   = alloc((size_t)BATCH * TSEQ * DIN);
  float* WTf = alloc((size_t)DIN * G4);
  float* WTb = alloc((size_t)DIN * G4);
  float* XgF = alloc((size_t)BATCH * TSEQ * G4);
  float* XgB = alloc((size_t)BATCH * TSEQ * G4);
  float* oF  = alloc((size_t)BATCH * TSEQ * HID);
  float* oB  = alloc((size_t)BATCH * TSEQ * HID);
  float* ctx = alloc((size_t)BATCH * 256);
  (void)ws_size; (void)n_in; (void)in_sizes; (void)out_size;

  // GCN layer 1
  k_init_deg<<<(N_NODES + 255) / 256, 256, 0, stream>>>(deg);
  k_deg_scatter<<<(N_EDGES + 255) / 256, 256, 0, stream>>>(ei, ew, deg);
  k_dis<<<(N_NODES + 255) / 256, 256, 0, stream>>>(deg, dis);
  k_xw1<<<(N_NODES * GHID) / 256, 256, 0, stream>>>(xc, emb, tf, W1, xw1);
  k_agg_init<<<(N_NODES * GHID) / 256, 256, 0, stream>>>(dis, xw1, b1, o1, 6, N_NODES * GHID);
  k_agg_scatter<<<(N_EDGES * GHID) / 256, 256, 0, stream>>>(ei, ew, dis, xw1, o1, 6);
  // GCN layer 2 (relu folded into xw2 read)
  k_xw2<<<(N_NODES * EMBD) / 256, 256, 0, stream>>>(o1, W2, xw2);
  k_agg_init<<<(N_NODES * EMBD) / 256, 256, 0, stream>>>(dis, xw2, b2, z, 7, N_NODES * EMBD);
  k_agg_scatter<<<(N_EDGES * EMBD) / 256, 256, 0, stream>>>(ei, ew, dis, xw2, z, 7);
  // sequence features
  k_build_x<<<BATCH * TSEQ, 160, 0, stream>>>(z, seq, dur, dW1, db1, dW2, db2, X);
  // input projections (WMMA)
  k_transpose_wih<<<(G4 * DIN + 255) / 256, 256, 0, stream>>>(WihF, WTf);
  k_transpose_wih<<<(G4 * DIN + 255) / 256, 256, 0, stream>>>(WihB, WTb);
  k_xg<<<dim3(4, (BATCH * TSEQ) / 16), 256, 0, stream>>>(X, WTf, bF, XgF);
  k_xg<<<dim3(4, (BATCH * TSEQ) / 16), 256, 0, stream>>>(X, WTb, bB, XgB);
  // recurrence (WMMA, Whh resident in 320KB LDS)
  k_lstm<<<BATCH / 16, 256, LSTM_LDS_FLOATS * sizeof(float), stream>>>(XgF, WhhF, lens, oF, 0);
  k_lstm<<<BATCH / 16, 256, LSTM_LDS_FLOATS * sizeof(float), stream>>>(XgB, WhhB, lens, oB, 1);
  // attention + FC (WMMA)
  k_attn<<<BATCH, 256, 0, stream>>>(oF, oB, attW, attB, lens, ctx);
  k_fc<<<dim3((N_NODES / 16 + 7) / 8, 16), 256, 0, stream>>>(ctx, fcW, fcb, out);
}